// SimpleRGCN_62474594287730
// MI455X (gfx1250) — compile-verified
//
#include <hip/hip_runtime.h>
#include <hip/hip_bf16.h>

typedef __attribute__((ext_vector_type(2))) float v2f;
typedef __attribute__((ext_vector_type(4))) float v4f;
typedef __attribute__((ext_vector_type(8))) float v8f;

#define F_IN 128   // K dimension (both layers)
#define MT   64    // rows per block
#define NT   64    // cols per block
#define NREL 3

// ---------------------------------------------------------------------------
// Fused GEMM: Y[n, c] for c in [0, C) where
//   c <  H           : (X @ Wroot)[n, c]
//   c >= H, block r  : (X @ Wrel[r])[n, c - H - r*H]
// V_WMMA_F32_16X16X4_F32 (fp32 exact; GEMM is ~5% of runtime, scatter is the
// bottleneck, so fp32 tensor ops are the right precision).
// Block: 128 threads = 4 wave32. LDS: 64x128 A tile + swizzled 128x64 B = 64KB.
// A tile staged with GLOBAL_LOAD_ASYNC_TO_LDS_B128 (ASYNCcnt-tracked DMA).
// ---------------------------------------------------------------------------
__global__ __launch_bounds__(128)
void rgcn_gemm_kernel(const float* __restrict__ X,
                      const float* __restrict__ Wroot,
                      const float* __restrict__ Wrel,
                      float* __restrict__ Y,
                      int Nrows, int H, int C)
{
    __shared__ float Alds[MT * F_IN];   // 32 KB, row-major 64 x 128
    __shared__ float Bsw[64 * 128];     // 32 KB, K-pair swizzle: [k/2][n*2 + (k&1)]

    const int tid  = threadIdx.x;       // 0..127
    const int lane = tid & 31;
    const int wave = tid >> 5;          // 0..3
    const int m0   = blockIdx.x * MT;
    const int c0   = blockIdx.y * NT;

    // ---- stage A tile (MT x 128): async DMA global -> LDS, b128 per lane ----
    // Garbage in OOB rows is harmless: it only feeds D rows that are never
    // stored (guarded by gr < Nrows).
    #pragma unroll
    for (int i = 0; i < (MT * F_IN / 4) / 128; ++i) {   // 16 iters
        int v   = tid + i * 128;
        int row = v >> 5;               // 32 float4 per 128-wide row
        int c4  = (v & 31) << 2;
        if (m0 + row < Nrows) {
            unsigned     ldsOff = (unsigned)(size_t)(Alds + row * F_IN + c4);
            const float* gptr   = X + (size_t)(m0 + row) * F_IN + c4;
            asm volatile("global_load_async_to_lds_b128 %0, %1, off"
                         :: "v"(ldsOff), "v"(gptr) : "memory");
        }
    }

    // ---- stage B panel (128 x NT) swizzled; NT=64 never crosses a block ----
    const float* Wp;
    int jbase;
    if (c0 < H) { Wp = Wroot; jbase = c0; }
    else {
        int r = (c0 - H) / H;
        jbase = (c0 - H) - r * H;
        Wp = Wrel + (size_t)r * F_IN * H;
    }
    // 64 K-pairs x 64 cols = 4096 float2 writes; 32 per thread.
    #pragma unroll
    for (int i = 0; i < (64 * NT) / 128; ++i) {         // 32 iters
        int v = tid + i * 128;
        int p = v >> 6;                 // K pair index 0..63
        int n = v & 63;                 // column within panel
        v2f w;
        w.x = Wp[(size_t)(2 * p)     * H + jbase + n];
        w.y = Wp[(size_t)(2 * p + 1) * H + jbase + n];
        *(v2f*)(Bsw + p * 128 + n * 2) = w;             // ds_store_b64
    }

    asm volatile("s_wait_asynccnt 0" ::: "memory");
    __syncthreads();

    // ---- WMMA main loop: wave 'w' owns M-tile w; 4 N-tiles of 16 ----
    const int mtile = wave;
    const int arow  = mtile * 16 + (lane & 15);
    const int ksel  = lane >> 4;             // 0 (lanes 0-15) or 1 (lanes 16-31)
    const int nb    = lane & 15;

    const float* Abase = Alds + arow * F_IN + (ksel << 1);
    const float* Bbase = Bsw + ksel * 128 + nb * 2;

    v8f acc0 = {}, acc1 = {}, acc2 = {}, acc3 = {};

    #pragma unroll 4
    for (int kt = 0; kt < F_IN / 4; ++kt) {
        // A frag (16x4 f32): VGPR0 = {K0 | K2}, VGPR1 = {K1 | K3}  (ISA 7.12.2)
        v2f a = *(const v2f*)(Abase + kt * 4);
        // B frag: contiguous {B[kr][n], B[kr+1][n]} thanks to the K-pair swizzle
        const float* bk = Bbase + kt * 256;              // kt*2 pairs * 128
        v2f b0 = *(const v2f*)(bk + 0);
        v2f b1 = *(const v2f*)(bk + 32);
        v2f b2 = *(const v2f*)(bk + 64);
        v2f b3 = *(const v2f*)(bk + 96);
        acc0 = __builtin_amdgcn_wmma_f32_16x16x4_f32(false, a, false, b0, (short)0, acc0, false, false);
        acc1 = __builtin_amdgcn_wmma_f32_16x16x4_f32(false, a, false, b1, (short)0, acc1, false, false);
        acc2 = __builtin_amdgcn_wmma_f32_16x16x4_f32(false, a, false, b2, (short)0, acc2, false, false);
        acc3 = __builtin_amdgcn_wmma_f32_16x16x4_f32(false, a, false, b3, (short)0, acc3, false, false);
    }

    // ---- store C/D: VGPR v -> row v (lanes 0-15) / v+8 (lanes 16-31) ----
    const int mbase = m0 + mtile * 16 + ((lane >> 4) << 3);
    const int ncol0 = c0 + nb;

#define STORE_ACC(ACC, NTI)                                                  \
    do {                                                                     \
        const int gc = ncol0 + (NTI) * 16;                                   \
        _Pragma("unroll")                                                    \
        for (int v = 0; v < 8; ++v) {                                        \
            const int gr = mbase + v;                                        \
            if (gr < Nrows) Y[(size_t)gr * C + gc] = (ACC)[v];               \
        }                                                                    \
    } while (0)

    STORE_ACC(acc0, 0);
    STORE_ACC(acc1, 1);
    STORE_ACC(acc2, 2);
    STORE_ACC(acc3, 3);
#undef STORE_ACC
}

// ---------------------------------------------------------------------------
// Edge pass: one wave32 per edge. Gather h_r[src] (H floats, vectorized),
// scatter-add into per-relation sum S[dst, r*H+j] and degree CNT[dst, r].
// Single pass over all edges / relations (edge lists read once per layer).
// ---------------------------------------------------------------------------
__global__ __launch_bounds__(256)
void rgcn_edge_kernel(const int* __restrict__ edge_index,   // [2, E]
                      const int* __restrict__ edge_type,    // [E]
                      const float* __restrict__ Y,          // [N, C]
                      float* __restrict__ S,                // [N, NREL*H]
                      float* __restrict__ CNT,              // [N, NREL]
                      int E, int H, int C)
{
    const int gw   = (int)((blockIdx.x * blockDim.x + threadIdx.x) >> 5);
    const int lane = threadIdx.x & 31;
    if (gw >= E) return;

    const int src = edge_index[gw];
    const int dst = edge_index[E + gw];
    const int r   = edge_type[gw];

    const float* hrow = Y + (size_t)src * C + H + (size_t)r * H;  // skip root block
    float*       srow = S + (size_t)dst * (NREL * H) + (size_t)r * H;

    if (H == 128) {
        const v4f h = *(const v4f*)(hrow + lane * 4);    // coalesced 512B/wave
        float* sp = srow + lane * 4;
        atomicAdd(sp + 0, h.x);
        atomicAdd(sp + 1, h.y);
        atomicAdd(sp + 2, h.z);
        atomicAdd(sp + 3, h.w);
    } else {                                             // H == 64
        const v2f h = *(const v2f*)(hrow + lane * 2);
        float* sp = srow + lane * 2;
        atomicAdd(sp + 0, h.x);
        atomicAdd(sp + 1, h.y);
    }
    if (lane == 0)
        atomicAdd(CNT + (size_t)dst * NREL + r, 1.0f);
}

// ---------------------------------------------------------------------------
// Combine: out[n,j] = rootY + bias + sum_r S_r / max(cnt_r, 1)  (opt. ReLU)
// ---------------------------------------------------------------------------
__global__ __launch_bounds__(256)
void rgcn_combine_kernel(const float* __restrict__ Y,
                         const float* __restrict__ S,
                         const float* __restrict__ CNT,
                         const float* __restrict__ bias,
                         float* __restrict__ Out,
                         int Nrows, int hshift, int C, int do_relu)
{
    const int H = 1 << hshift;
    const long long idx = (long long)blockIdx.x * blockDim.x + threadIdx.x;
    if (idx >= ((long long)Nrows << hshift)) return;
    const int n = (int)(idx >> hshift);
    const int j = (int)(idx & (H - 1));

    float acc = Y[(size_t)n * C + j] + bias[j];
    #pragma unroll
    for (int r = 0; r < NREL; ++r) {
        const float c = CNT[(size_t)n * NREL + r];
        acc += S[(size_t)n * (NREL * H) + (size_t)r * H + j] / fmaxf(c, 1.0f);
    }
    if (do_relu) acc = fmaxf(acc, 0.0f);
    Out[idx] = acc;
}

__global__ __launch_bounds__(256)
void zero_kernel(float* __restrict__ p, long long n)
{
    const long long i = (long long)blockIdx.x * blockDim.x + threadIdx.x;
    if (i < n) p[i] = 0.0f;
}

// ---------------------------------------------------------------------------
extern "C" void kernel_launch(void* const* d_in, const int* in_sizes, int n_in,
                              void* d_out, int out_size, void* d_ws, size_t ws_size,
                              hipStream_t stream)
{
    const float* x          = (const float*)d_in[0];
    const int*   edge_index = (const int*)  d_in[1];
    const int*   edge_type  = (const int*)  d_in[2];
    const float* W1         = (const float*)d_in[3];  // [3,128,128]
    const float* root1      = (const float*)d_in[4];  // [128,128]
    const float* b1         = (const float*)d_in[5];  // [128]
    const float* W2         = (const float*)d_in[6];  // [3,128,64]
    const float* root2      = (const float*)d_in[7];  // [128,64]
    const float* b2         = (const float*)d_in[8];  // [64]
    float*       out        = (float*)d_out;

    const int N = in_sizes[0] / F_IN;     // 50000
    const int E = in_sizes[2];            // 800000

    // Workspace layout (floats). Layer 2 reuses Y/S/CNT regions of layer 1.
    float* Y1  = (float*)d_ws;                 // N * 512  (root | 3 relations)
    float* S1  = Y1 + (size_t)N * 512;         // N * 384
    float* C1  = S1 + (size_t)N * 384;         // N * 3
    float* H1  = C1 + (size_t)N * 3;           // N * 128
    float* Y2  = Y1;                           // N * 256
    float* S2  = S1;                           // N * 192
    float* C2  = C1;                           // N * 3

    const long long zcount = (long long)N * 384 + (long long)N * 3; // S+CNT region
    const long long ethreads = (long long)E * 32;

    // ------------------ Layer 1 (H = 128, C = 512) ------------------
    {
        dim3 grid((N + MT - 1) / MT, 512 / NT);
        rgcn_gemm_kernel<<<grid, 128, 0, stream>>>(x, root1, W1, Y1, N, 128, 512);
    }
    zero_kernel<<<(unsigned)((zcount + 255) / 256), 256, 0, stream>>>(S1, zcount);
    rgcn_edge_kernel<<<(unsigned)((ethreads + 255) / 256), 256, 0, stream>>>(
        edge_index, edge_type, Y1, S1, C1, E, 128, 512);
    {
        const long long t = (long long)N * 128;
        rgcn_combine_kernel<<<(unsigned)((t + 255) / 256), 256, 0, stream>>>(
            Y1, S1, C1, b1, H1, N, /*hshift=*/7, 512, /*relu=*/1);
    }

    // ------------------ Layer 2 (H = 64, C = 256) -------------------
    {
        dim3 grid((N + MT - 1) / MT, 256 / NT);
        rgcn_gemm_kernel<<<grid, 128, 0, stream>>>(H1, root2, W2, Y2, N, 64, 256);
    }
    zero_kernel<<<(unsigned)((zcount + 255) / 256), 256, 0, stream>>>(S2, zcount);
    rgcn_edge_kernel<<<(unsigned)((ethreads + 255) / 256), 256, 0, stream>>>(
        edge_index, edge_type, Y2, S2, C2, E, 64, 256);
    {
        const long long t = (long long)N * 64;
        rgcn_combine_kernel<<<(unsigned)((t + 255) / 256), 256, 0, stream>>>(
            Y2, S2, C2, b2, out, N, /*hshift=*/6, 256, /*relu=*/0);
    }
}